// MTLoss_47802986005050
// MI455X (gfx1250) — compile-verified
//
#include <hip/hip_runtime.h>
#include <stdint.h>

#define N_TOTAL   279424
#define NCLS      21
#define BATCH     32
#define MAXB      100
#define SEG_HW    300
#define SEG_IMG   (SEG_HW*SEG_HW)          // 90000
#define SEG_PIX   (BATCH*SEG_IMG)          // 2,880,000
#define CLS_ELEMS (N_TOTAL*NCLS)           // 5,867,904 (divisible by 4)
#define NBLK_CLS  ((N_TOTAL + 255) / 256)  // 1092
#define SEG_PPB   512                      // pixels per block (2 per thread)
#define NBLK_SEG  (SEG_PIX / SEG_PPB)      // 5625 (exact)
#define CLS_CHUNKS (256 * NCLS / 4)        // 1344 16-byte chunks per cls block

typedef float v2f __attribute__((ext_vector_type(2)));
typedef float v8f __attribute__((ext_vector_type(8)));

// ---------------------------------------------------------------------------
// CDNA5 async global->LDS staging (ASYNCcnt tracked), non-temporal policy:
// Seg (242MB) > L2 (192MB) and is single-use -> NT avoids L2 thrash.
// ---------------------------------------------------------------------------
__device__ __forceinline__ void async_load_b32(uint32_t lds_byte_off, const float* gaddr) {
    asm volatile("global_load_async_to_lds_b32 %0, %1, off th:TH_LOAD_NT"
                 :: "v"(lds_byte_off), "v"(gaddr) : "memory");
}
__device__ __forceinline__ void async_load_b64(uint32_t lds_byte_off, const float* gaddr) {
    asm volatile("global_load_async_to_lds_b64 %0, %1, off th:TH_LOAD_NT"
                 :: "v"(lds_byte_off), "v"(gaddr) : "memory");
}
__device__ __forceinline__ void async_load_b128(uint32_t lds_byte_off, const float4* gaddr) {
    asm volatile("global_load_async_to_lds_b128 %0, %1, off th:TH_LOAD_NT"
                 :: "v"(lds_byte_off), "v"(gaddr) : "memory");
}
__device__ __forceinline__ void async_wait0() {
    asm volatile("s_wait_asynccnt 0" ::: "memory");
}

// ---------------------------------------------------------------------------
// Wave32 sum via V_WMMA_F32_16X16X4_F32 (EXEC must be all-ones):
//   A(16x4): lanes0-15 -> {K0=v,K1=0}, lanes16-31 -> {K2=v,K3=0}; B = ones
//   D[m][n] = v[m] + v[m+16]; 8 VGPR adds give half-sums, shfl_xor(16) folds.
// ---------------------------------------------------------------------------
__device__ __forceinline__ float wave_sum32(float v) {
    v2f a; a[0] = v;    a[1] = 0.0f;
    v2f b; b[0] = 1.0f; b[1] = 1.0f;
    v8f c = {0.f,0.f,0.f,0.f,0.f,0.f,0.f,0.f};
    v8f d = __builtin_amdgcn_wmma_f32_16x16x4_f32(false, a, false, b, (short)0, c, false, false);
    float s = d[0]+d[1]+d[2]+d[3]+d[4]+d[5]+d[6]+d[7];
    s += __shfl_xor(s, 16, 32);
    return s;
}

// Block (256 = 8 waves) sum; result valid on all threads.
__device__ __forceinline__ float block_sum(float v, float* red8) {
    float w = wave_sum32(v);
    int lane = threadIdx.x & 31, wid = threadIdx.x >> 5;
    __syncthreads();                 // protect red8 against previous use
    if (lane == 0) red8[wid] = w;
    __syncthreads();
    return red8[0]+red8[1]+red8[2]+red8[3]+red8[4]+red8[5]+red8[6]+red8[7];
}

// ---------------------------------------------------------------------------
// 1) init targets: ClsT = fill (0 or -1 by mining), LocT = 0
// ---------------------------------------------------------------------------
__global__ __launch_bounds__(256) void k_init(int* __restrict__ ClsT, float* __restrict__ LocT,
                                              const int* __restrict__ mining) {
    int i = blockIdx.x * 256 + threadIdx.x;
    int fill = (mining[0] != 0) ? -1 : 0;
    if (i < N_TOTAL) {
        ClsT[i] = fill;
        float4 z = {0.f,0.f,0.f,0.f};
        ((float4*)LocT)[i] = z;
    }
}

// ---------------------------------------------------------------------------
// 2) scatter 32x100 boxes into flat target arrays
// ---------------------------------------------------------------------------
__global__ __launch_bounds__(256) void k_scatter(const float* __restrict__ gt, const float* __restrict__ df,
                                                 const int* __restrict__ idx, const int* __restrict__ clsb,
                                                 int* __restrict__ ClsT, float* __restrict__ LocT) {
    int t = blockIdx.x * 256 + threadIdx.x;
    if (t >= BATCH * MAXB) return;
    const int MAPS[6] = {38,19,10,5,3,1};
    const int NBX[6]  = {4,6,6,6,4,4};
    const int BASE[6] = {0,184832,254144,273344,278144,279296};
    int b = t / MAXB;
    const int* e = idx + t * 4;
    int layer = e[1], pos = e[2], boxn = e[3];
    bool valid = (layer != -100);
    int lc = layer < 0 ? 0 : (layer > 5 ? 5 : layer);
    int H = MAPS[lc], nb = NBX[lc], base = BASE[lc];
    int r = pos / H, c = pos % H;
    int flat = base + (((b * H + r) * H + c) * nb + boxn);
    if (valid && flat >= 0 && flat < N_TOTAL) {
        ClsT[flat] = clsb[t];
        float4 g = ((const float4*)gt)[t];
        float4 d = ((const float4*)df)[t];
        float4 o;
        o.x = (g.x - d.x) / 0.1f;  o.y = (g.y - d.y) / 0.1f;
        o.z = (g.z - d.z) / 0.1f;  o.w = (g.w - d.w) / 0.1f;
        ((float4*)LocT)[flat] = o;
    }
}

// ---------------------------------------------------------------------------
// 3) cls CE + loc smooth-L1; 256 anchors/block, 21KB slab staged with 6 b128
//    asyncs per thread (fully coalesced 16B lanes)
// ---------------------------------------------------------------------------
__global__ __launch_bounds__(256) void k_cls(const float* __restrict__ Cls, const float* __restrict__ Loc,
                      const int* __restrict__ ClsT, const float* __restrict__ LocT,
                      float* __restrict__ p_cls, float* __restrict__ p_cnt,
                      float* __restrict__ p_loc, float* __restrict__ p_npos) {
    __shared__ __align__(16) float stage[256 * NCLS];   // 21504 B = 1344 x 16B chunks
    __shared__ float red8[8];
    int tid = threadIdx.x;
    long chunkBase = (long)blockIdx.x * CLS_CHUNKS;
    const float4* g4 = (const float4*)Cls;
    uint32_t lbase = (uint32_t)(uintptr_t)stage;
    #pragma unroll
    for (int j = 0; j < 6; ++j) {                       // 6*256 = 1536 covers 1344 chunks
        int ch = j * 256 + tid;
        if (ch > CLS_CHUNKS - 1) ch = CLS_CHUNKS - 1;   // clamp: EXEC stays full, same data
        long gch = chunkBase + ch;
        if (gch > (long)(CLS_ELEMS / 4) - 1) gch = (long)(CLS_ELEMS / 4) - 1;
        async_load_b128(lbase + 16u * (uint32_t)ch, g4 + gch);
    }
    async_wait0();
    __syncthreads();                                    // rows were loaded by other lanes

    int i = blockIdx.x * 256 + tid;
    float nll = 0.f, cnt = 0.f, lsum = 0.f, npos = 0.f;
    if (i < N_TOTAL) {
        const float* row = stage + tid * NCLS;          // stride 21: bank-conflict free
        float m = row[0];
        #pragma unroll
        for (int k = 1; k < NCLS; ++k) m = fmaxf(m, row[k]);
        float s = 0.f;
        #pragma unroll
        for (int k = 0; k < NCLS; ++k) s += __expf(row[k] - m);
        float lse = m + __logf(s);
        int t = ClsT[i];
        if (t != -1) {
            int safe = t < 0 ? 0 : (t > NCLS - 1 ? NCLS - 1 : t);
            nll = lse - row[safe];
            cnt = 1.f;
        }
        if (t > 0) {
            float4 l  = ((const float4*)Loc)[i];
            float4 lt = ((const float4*)LocT)[i];
            float sl = 0.f, d;
            d = fabsf(l.x - lt.x); sl += (d < 1.f) ? 0.5f * d * d : d - 0.5f;
            d = fabsf(l.y - lt.y); sl += (d < 1.f) ? 0.5f * d * d : d - 0.5f;
            d = fabsf(l.z - lt.z); sl += (d < 1.f) ? 0.5f * d * d : d - 0.5f;
            d = fabsf(l.w - lt.w); sl += (d < 1.f) ? 0.5f * d * d : d - 0.5f;
            lsum = sl; npos = 1.f;
        }
    }
    float bs;
    bs = block_sum(nll,  red8); if (tid == 0) p_cls [blockIdx.x] = bs;
    bs = block_sum(cnt,  red8); if (tid == 0) p_cnt [blockIdx.x] = bs;
    bs = block_sum(lsum, red8); if (tid == 0) p_loc [blockIdx.x] = bs;
    bs = block_sum(npos, red8); if (tid == 0) p_npos[blockIdx.x] = bs;
}

// ---------------------------------------------------------------------------
// 4) seg CE; 512 pixels/block (2 per thread), 21 class planes staged with b64
//    asyncs (pairs never straddle an image: 90000 is even)
// ---------------------------------------------------------------------------
__global__ __launch_bounds__(256) void k_seg(const float* __restrict__ Seg, const int* __restrict__ lbl,
                                             float* __restrict__ p_sum, float* __restrict__ p_cnt) {
    __shared__ __align__(16) float stage[SEG_PPB * NCLS];   // 43008 B
    __shared__ float red8[8];
    int tid = threadIdx.x;
    int p0 = blockIdx.x * SEG_PPB + 2 * tid;
    int b = p0 / SEG_IMG;
    int hw = p0 - b * SEG_IMG;
    const float* basep = Seg + (long)b * NCLS * SEG_IMG + hw;
    uint32_t lbase = (uint32_t)(uintptr_t)stage;
    #pragma unroll
    for (int c = 0; c < NCLS; ++c)                          // 21 in-flight b64 asyncs
        async_load_b64(lbase + 4u * (uint32_t)(c * SEG_PPB + 2 * tid),
                       basep + (long)c * SEG_IMG);
    async_wait0();                                          // each thread reads only its own slots

    float nll = 0.f, cnt = 0.f;
    #pragma unroll
    for (int q = 0; q < 2; ++q) {
        int sl0 = 2 * tid + q;
        float m = stage[sl0];
        #pragma unroll
        for (int c = 1; c < NCLS; ++c) m = fmaxf(m, stage[c * SEG_PPB + sl0]);
        float s = 0.f;
        #pragma unroll
        for (int c = 0; c < NCLS; ++c) s += __expf(stage[c * SEG_PPB + sl0] - m);
        float lse = m + __logf(s);
        int t = lbl[p0 + q];
        if (t != -1) {
            int safe = t < 0 ? 0 : (t > NCLS - 1 ? NCLS - 1 : t);
            nll += lse - stage[safe * SEG_PPB + sl0];
            cnt += 1.f;
        }
    }
    float bs;
    bs = block_sum(nll, red8); if (tid == 0) p_sum[blockIdx.x] = bs;
    bs = block_sum(cnt, red8); if (tid == 0) p_cnt[blockIdx.x] = bs;
}

// ---------------------------------------------------------------------------
// 5) deterministic final reduce of per-block partials -> 3 scalars
// ---------------------------------------------------------------------------
__global__ __launch_bounds__(256) void k_final(const float* __restrict__ p_cls, const float* __restrict__ p_cnt,
                        const float* __restrict__ p_loc, const float* __restrict__ p_npos,
                        const float* __restrict__ p_seg, const float* __restrict__ p_segcnt,
                        float* __restrict__ out) {
    __shared__ float red8[8];
    int tid = threadIdx.x;
    float a = 0.f, b = 0.f, c = 0.f, d = 0.f, e = 0.f, f = 0.f;
    for (int i = tid; i < NBLK_CLS; i += 256) { a += p_cls[i]; b += p_cnt[i]; c += p_loc[i]; d += p_npos[i]; }
    for (int i = tid; i < NBLK_SEG; i += 256) { e += p_seg[i]; f += p_segcnt[i]; }
    a = block_sum(a, red8);  b = block_sum(b, red8);
    c = block_sum(c, red8);  d = block_sum(d, red8);
    e = block_sum(e, red8);  f = block_sum(f, red8);
    if (tid == 0) {
        out[0] = a / fmaxf(b, 1.f);   // loss_cls
        out[1] = c / d;               // loss_loc (reference divides by n_pos directly)
        out[2] = e / fmaxf(f, 1.f);   // loss_seg
    }
}

extern "C" void kernel_launch(void* const* d_in, const int* in_sizes, int n_in,
                              void* d_out, int out_size, void* d_ws, size_t ws_size,
                              hipStream_t stream) {
    const float* Loc  = (const float*)d_in[0];
    const float* Cls  = (const float*)d_in[1];
    const float* Seg  = (const float*)d_in[2];
    const float* gt   = (const float*)d_in[3];
    const float* df   = (const float*)d_in[4];
    const int*   idx  = (const int*)  d_in[5];
    const int*   clsb = (const int*)  d_in[6];
    const int*   mining = (const int*)d_in[8];
    const int*   slbl = (const int*)  d_in[9];
    float* out = (float*)d_out;

    // workspace layout
    char* ws = (char*)d_ws;
    int*   ClsT = (int*)  ws;                                  ws += (size_t)N_TOTAL * 4;
    float* LocT = (float*)ws;                                  ws += (size_t)N_TOTAL * 16;
    float* p_cls    = (float*)ws;                              ws += (size_t)NBLK_CLS * 4;
    float* p_cnt    = (float*)ws;                              ws += (size_t)NBLK_CLS * 4;
    float* p_loc    = (float*)ws;                              ws += (size_t)NBLK_CLS * 4;
    float* p_npos   = (float*)ws;                              ws += (size_t)NBLK_CLS * 4;
    float* p_seg    = (float*)ws;                              ws += (size_t)NBLK_SEG * 4;
    float* p_segcnt = (float*)ws;

    k_init   <<<NBLK_CLS, 256, 0, stream>>>(ClsT, LocT, mining);
    k_scatter<<<(BATCH * MAXB + 255) / 256, 256, 0, stream>>>(gt, df, idx, clsb, ClsT, LocT);
    k_cls    <<<NBLK_CLS, 256, 0, stream>>>(Cls, Loc, ClsT, LocT, p_cls, p_cnt, p_loc, p_npos);
    k_seg    <<<NBLK_SEG, 256, 0, stream>>>(Seg, slbl, p_seg, p_segcnt);
    k_final  <<<1, 256, 0, stream>>>(p_cls, p_cnt, p_loc, p_npos, p_seg, p_segcnt, out);
}